// PlaylistEmbedding_44779329028609
// MI455X (gfx1250) — compile-verified
//
#include <hip/hip_runtime.h>

// PlaylistEmbedding (embedding-bag sum) for MI455X / gfx1250.
//
// out[b, :] = sum_{l<HIST} w[idx[b,l], :] + bias      (EMBED = 32 fp32)
//
// Memory-bound gather kernel: ~419 MB of 128B row gathers (table is 10.4 MB ->
// resident in 192MB L2), 26 MB of streamed int64 indices, 2 MB stores.
// CDNA5-specific paths: global_load_async_to_lds_b64 (+ s_wait_asynccnt) to
// stage indices in LDS; b128 coalesced gathers (4 groups x 8 lanes x float4
// per wave32). HIST=200 is specialized at compile time for static loop
// control and 4-deep gather MLP.

typedef float f4 __attribute__((ext_vector_type(4)));
typedef int   v2i __attribute__((vector_size(8)));   // matches async-LDS builtin param type

#define WAVES_PER_BLOCK 8
#define MAX_HIST 200   // reference HIST; LDS staging sized for this
#define EMBED 32

__device__ __forceinline__ void stage_indices_async(const long long* gsrc,
                                                    long long* lbase,
                                                    int lane, int hist)
{
#if __has_builtin(__builtin_amdgcn_global_load_async_to_lds_b64)
    for (int t = lane; t < hist; t += 32) {
        __builtin_amdgcn_global_load_async_to_lds_b64(
            (__attribute__((address_space(1))) v2i*)(gsrc + t),
            (__attribute__((address_space(3))) v2i*)(lbase + t),
            /*imm offset*/0, /*cpol*/0);
    }
  #if __has_builtin(__builtin_amdgcn_s_wait_asynccnt)
    __builtin_amdgcn_s_wait_asynccnt(0);
  #else
    asm volatile("s_wait_asynccnt 0" ::: "memory");
  #endif
#else
    for (int t = lane; t < hist; t += 32) lbase[t] = gsrc[t];
#endif
}

__device__ __forceinline__ f4 gather_row(const float* __restrict__ w, int row, int c)
{
    return ((const f4*)(w + (size_t)(unsigned)row * EMBED))[c];
}

__device__ __forceinline__ void reduce_and_store(f4 acc,
                                                 const float* __restrict__ bias,
                                                 float* __restrict__ out,
                                                 int b, int lane, int c)
{
    // Fold the 4 group partials (wave32 butterfly over lane bits 3,4).
    for (int m = 8; m < 32; m <<= 1) {
        acc.x += __shfl_xor(acc.x, m, 32);
        acc.y += __shfl_xor(acc.y, m, 32);
        acc.z += __shfl_xor(acc.z, m, 32);
        acc.w += __shfl_xor(acc.w, m, 32);
    }
    if (lane < 8) {   // lanes 0..7: bias + one coalesced 128B row store
        f4 r = acc + ((const f4*)bias)[c];
        ((f4*)(out + (size_t)b * EMBED))[c] = r;
    }
}

// ---- Compile-time HIST specialization (reference: HIST=200) --------------
template <int HIST>
__global__ __launch_bounds__(256) void playlist_embed_bag_fixed(
    const long long* __restrict__ idx,
    const float*     __restrict__ w,
    const float*     __restrict__ bias,
    float*           __restrict__ out,
    int batch)
{
    __shared__ long long s_idx[WAVES_PER_BLOCK * HIST];

    const int lane = threadIdx.x & 31;
    const int wv   = threadIdx.x >> 5;
    const int b    = blockIdx.x * WAVES_PER_BLOCK + wv;
    if (b >= batch) return;                      // wave-uniform exit

    const long long* gsrc  = idx + (size_t)b * HIST;
    long long*       lbase = &s_idx[wv * HIST];
    stage_indices_async(gsrc, lbase, lane, HIST);

    // 4 groups of 8 lanes; group g handles entry (l+g); lane c=lane&7 owns
    // embedding columns [4c,4c+3] -> one global_load_b128 per entry.
    const int* li = (const int*)lbase;           // low dword of each int64 (LE)
    const int  g  = lane >> 3;
    const int  c  = lane & 7;

    f4 acc0 = {0.f,0.f,0.f,0.f}, acc1 = {0.f,0.f,0.f,0.f};
    f4 acc2 = {0.f,0.f,0.f,0.f}, acc3 = {0.f,0.f,0.f,0.f};

    constexpr int MAIN = (HIST / 16) * 16;       // 192 for HIST=200
    for (int l = 0; l < MAIN; l += 16) {         // 4 independent gathers/iter
        int r0 = li[2 * (l +      g)];
        int r1 = li[2 * (l +  4 + g)];
        int r2 = li[2 * (l +  8 + g)];
        int r3 = li[2 * (l + 12 + g)];
        acc0 += gather_row(w, r0, c);
        acc1 += gather_row(w, r1, c);
        acc2 += gather_row(w, r2, c);
        acc3 += gather_row(w, r3, c);
    }
#pragma unroll
    for (int l = MAIN; l + 4 <= HIST; l += 4) {  // static remainder (8 for 200)
        int r0 = li[2 * (l + g)];
        acc0 += gather_row(w, r0, c);
    }
    if constexpr ((HIST & 3) != 0) {             // not hit for HIST=200
        int l = (HIST / 4) * 4;
        if (l + g < HIST) acc0 += gather_row(w, li[2 * (l + g)], c);
    }

    reduce_and_store((acc0 + acc1) + (acc2 + acc3), bias, out, b, lane, c);
}

// ---- Generic fallback (runtime hist <= MAX_HIST) -------------------------
__global__ __launch_bounds__(256) void playlist_embed_bag_generic(
    const long long* __restrict__ idx,
    const float*     __restrict__ w,
    const float*     __restrict__ bias,
    float*           __restrict__ out,
    int batch, int hist)
{
    __shared__ long long s_idx[WAVES_PER_BLOCK * MAX_HIST];

    const int lane = threadIdx.x & 31;
    const int wv   = threadIdx.x >> 5;
    const int b    = blockIdx.x * WAVES_PER_BLOCK + wv;
    if (b >= batch) return;

    const long long* gsrc  = idx + (size_t)b * hist;
    long long*       lbase = &s_idx[wv * MAX_HIST];
    stage_indices_async(gsrc, lbase, lane, hist);

    const int* li = (const int*)lbase;
    const int  g  = lane >> 3;
    const int  c  = lane & 7;

    f4 acc0 = {0.f,0.f,0.f,0.f}, acc1 = {0.f,0.f,0.f,0.f};
    int l = 0;
    for (; l + 8 <= hist; l += 8) {
        int r0 = li[2 * (l + g)];
        int r1 = li[2 * (l + 4 + g)];
        acc0 += gather_row(w, r0, c);
        acc1 += gather_row(w, r1, c);
    }
    for (; l + 4 <= hist; l += 4) acc0 += gather_row(w, li[2 * (l + g)], c);
    if (l + g < hist)             acc0 += gather_row(w, li[2 * (l + g)], c);

    reduce_and_store(acc0 + acc1, bias, out, b, lane, c);
}

extern "C" void kernel_launch(void* const* d_in, const int* in_sizes, int n_in,
                              void* d_out, int out_size, void* d_ws, size_t ws_size,
                              hipStream_t stream) {
    const long long* idx  = (const long long*)d_in[0];  // int64 per reference
    const float*     w    = (const float*)d_in[1];
    const float*     bias = (const float*)d_in[2];
    float*           out  = (float*)d_out;

    const int embed = in_sizes[2];            // 32
    const int batch = out_size / embed;       // 16384
    const int hist  = in_sizes[0] / batch;    // 200

    const int blocks = (batch + WAVES_PER_BLOCK - 1) / WAVES_PER_BLOCK;
    if (hist == MAX_HIST) {
        playlist_embed_bag_fixed<MAX_HIST><<<blocks, 256, 0, stream>>>(
            idx, w, bias, out, batch);
    } else {
        playlist_embed_bag_generic<<<blocks, 256, 0, stream>>>(
            idx, w, bias, out, batch, hist);
    }
}